// ProtocolTreeGAttention_46600395162031
// MI455X (gfx1250) — compile-verified
//
#include <hip/hip_runtime.h>
#include <hip/hip_bf16.h>

#define NN     50000
#define EE     800000
#define EPLUS  (EE + NN)       // edges + self loops
#define GRAPHS 512
#define NCLS   120

typedef float v2f __attribute__((ext_vector_type(2)));
typedef float v8f __attribute__((ext_vector_type(8)));

// ---------- order-preserving float<->uint map for atomic max ----------
__device__ __forceinline__ unsigned flipf(float f) {
  unsigned u = __float_as_uint(f);
  return (u & 0x80000000u) ? ~u : (u | 0x80000000u);
}
__device__ __forceinline__ float unflipf(unsigned e) {
  unsigned u = (e & 0x80000000u) ? (e ^ 0x80000000u) : ~e;
  return __uint_as_float(u);
}
#define ENC_NEG_INF 0x007FFFFFu   // flipf(-inf)

// ---------- generic fill ----------
__global__ void fill_u32_kernel(unsigned* __restrict__ p, unsigned v, int n) {
  int i = blockIdx.x * blockDim.x + threadIdx.x;
  if (i < n) p[i] = v;
}

// ---------- f32 WMMA GEMM: C[M,NCOL] = A[M,K] * B[K,NCOL] ----------
// One wave32 per 16x16 output tile; K stepped by 4 via V_WMMA_F32_16X16X4_F32.
// A frag (16x4): lane L holds rows M=L%16, K = (L/16)*2 + v  (v = vgpr 0..1)
// B frag (4x16): lane L holds col  N=L%16, K = (L/16)*2 + v
// C frag (16x16): vgpr r, lanes 0-15 -> M=r, lanes 16-31 -> M=r+8 (ISA 7.12.2)
template<int K, int NCOL>
__global__ void wmma_gemm_f32(const float* __restrict__ A, const float* __restrict__ B,
                              float* __restrict__ C) {
  const int gw   = (blockIdx.x * blockDim.x + threadIdx.x) >> 5;
  const int lane = threadIdx.x & 31;
  constexpr int TN = NCOL / 16;
  const int tm = gw / TN;
  const int tn = gw - tm * TN;
  const int half = lane >> 4;            // 0 or 1
  const int l16  = lane & 15;
  const int kl   = half * 2;
  const float* ap = A + (size_t)(tm * 16 + l16) * K + kl;
  const float* bp = B + (size_t)kl * NCOL + (tn * 16 + l16);
  v8f c = {};
#pragma unroll 4
  for (int kk = 0; kk < K; kk += 4) {
    v2f a = *(const v2f*)(ap + kk);      // two consecutive K values, 8B aligned
    v2f b;
    b.x = bp[(size_t)kk * NCOL];
    b.y = bp[(size_t)(kk + 1) * NCOL];
    c = __builtin_amdgcn_wmma_f32_16x16x4_f32(false, a, false, b, (short)0, c,
                                              false, false);
  }
  float* cp = C + (size_t)(tm * 16 + half * 8) * NCOL + (tn * 16 + l16);
#pragma unroll
  for (int r = 0; r < 8; ++r) cp[(size_t)r * NCOL] = c[r];
}

// ---------- attention projections: al_s[n,h], al_d[n,h] ----------
__global__ void attn_dots_kernel(const float* __restrict__ hf, const float* __restrict__ asrc,
                                 const float* __restrict__ adst, float* __restrict__ als,
                                 float* __restrict__ ald, int H) {
  int t = blockIdx.x * blockDim.x + threadIdx.x;
  if (t >= NN * H) return;
  int h = t % H;
  const float* hp = hf + (size_t)t * 64;   // [N,H,64] row-major -> (n*H+h)*64
  const float* as = asrc + h * 64;
  const float* ad = adst + h * 64;
  float s = 0.f, d = 0.f;
#pragma unroll 8
  for (int j = 0; j < 64; ++j) {
    float v = hp[j];
    s = fmaf(v, as[j], s);
    d = fmaf(v, ad[j], d);
  }
  als[t] = s;
  ald[t] = d;
}

__device__ __forceinline__ void edge_sd(const long long* __restrict__ ei, int e,
                                        int& s, int& d) {
  if (e < EE) { s = (int)ei[e]; d = (int)ei[EE + e]; }
  else        { s = d = e - EE; }
}

// ---------- pass A: segment max of leaky-relu logits over dst ----------
template<int H>
__global__ void edge_max_kernel(const float* __restrict__ als, const float* __restrict__ ald,
                                const long long* __restrict__ ei, unsigned* __restrict__ m) {
  int t = blockIdx.x * blockDim.x + threadIdx.x;
  if (t >= EPLUS * H) return;
  int e = t / H, h = t - e * H;
  int s, d; edge_sd(ei, e, s, d);
  float logit = als[s * H + h] + ald[d * H + h];
  logit = logit > 0.f ? logit : 0.2f * logit;
  atomicMax(&m[d * H + h], flipf(logit));
}

// ---------- pass B: exp weights + segment sum ----------
template<int H>
__global__ void edge_expsum_kernel(const float* __restrict__ als, const float* __restrict__ ald,
                                   const long long* __restrict__ ei, const unsigned* __restrict__ m,
                                   float* __restrict__ den, float* __restrict__ ew) {
  int t = blockIdx.x * blockDim.x + threadIdx.x;
  if (t >= EPLUS * H) return;
  int e = t / H, h = t - e * H;
  int s, d; edge_sd(ei, e, s, d);
  float logit = als[s * H + h] + ald[d * H + h];
  logit = logit > 0.f ? logit : 0.2f * logit;
  float w = expf(logit - unflipf(m[d * H + h]));
  ew[t] = w;
  atomicAdd(&den[d * H + h], w);
}

// ---------- pass C: weighted message aggregation (block per edge) ----------
template<int H, int D>
__global__ void edge_agg_kernel(const float* __restrict__ hf, const float* __restrict__ ew,
                                const float* __restrict__ den, const long long* __restrict__ ei,
                                float* __restrict__ out) {
  int e   = blockIdx.x;
  int idx = threadIdx.x;               // [0, H*D)
  int h   = idx / D;
  int s, d; edge_sd(ei, e, s, d);
  float alpha = ew[e * H + h] / (den[d * H + h] + 1e-16f);
  atomicAdd(&out[(size_t)d * (H * D) + idx], hf[(size_t)s * (H * D) + idx] * alpha);
}

// ---------- bias + ELU (layer-1 epilogue) ----------
__global__ void bias_elu_kernel(float* __restrict__ out, const float* __restrict__ b) {
  int i = blockIdx.x * blockDim.x + threadIdx.x;
  if (i >= NN * 256) return;
  float v = out[i] + b[i & 255];
  out[i] = v > 0.f ? v : (expf(v) - 1.f);
}

// ---------- global mean pool (sum + count); b2 added pre-pool per reference ----------
__global__ void pool_kernel(const float* __restrict__ out2, const float* __restrict__ b2,
                            const long long* __restrict__ batch, float* __restrict__ gsum,
                            float* __restrict__ gcnt) {
  int i = blockIdx.x * blockDim.x + threadIdx.x;
  if (i >= NN * 64) return;
  int n = i >> 6, d = i & 63;
  int g = (int)batch[n];
  atomicAdd(&gsum[g * 64 + d], out2[i] + b2[d]);
  if (d == 0) atomicAdd(&gcnt[g], 1.0f);
}

// ---------- classifier: [512,64] @ [64,120] + bc (tiny) ----------
__global__ void classifier_kernel(const float* __restrict__ gsum, const float* __restrict__ gcnt,
                                  const float* __restrict__ Wc, const float* __restrict__ bc,
                                  float* __restrict__ out) {
  int i = blockIdx.x * blockDim.x + threadIdx.x;
  if (i >= GRAPHS * NCLS) return;
  int g = i / NCLS, c = i - g * NCLS;
  float inv = 1.0f / fmaxf(gcnt[g], 1.0f);
  float acc = bc[c];
#pragma unroll 8
  for (int d = 0; d < 64; ++d)
    acc = fmaf(gsum[g * 64 + d] * inv, Wc[d * NCLS + c], acc);
  out[i] = acc;
}

static inline unsigned cdivll(long long a, long long b) { return (unsigned)((a + b - 1) / b); }

extern "C" void kernel_launch(void* const* d_in, const int* in_sizes, int n_in,
                              void* d_out, int out_size, void* d_ws, size_t ws_size,
                              hipStream_t stream) {
  (void)in_sizes; (void)n_in; (void)out_size; (void)ws_size;
  const float* x      = (const float*)d_in[0];
  const float* W1     = (const float*)d_in[1];
  const float* a_src1 = (const float*)d_in[2];
  const float* a_dst1 = (const float*)d_in[3];
  const float* b1     = (const float*)d_in[4];
  const float* W2     = (const float*)d_in[5];
  const float* a_src2 = (const float*)d_in[6];
  const float* a_dst2 = (const float*)d_in[7];
  const float* b2     = (const float*)d_in[8];
  const float* Wc     = (const float*)d_in[9];
  const float* bc     = (const float*)d_in[10];
  const long long* ei    = (const long long*)d_in[11];
  const long long* batch = (const long long*)d_in[12];

  // ---- workspace layout (floats) ----
  float* ws = (float*)d_ws;
  size_t o = 0;
  float*    h1   = ws + o; o += (size_t)NN * 256;
  float*    out1 = ws + o; o += (size_t)NN * 256;
  float*    h2   = ws + o; o += (size_t)NN * 64;
  float*    out2 = ws + o; o += (size_t)NN * 64;
  float*    als1 = ws + o; o += (size_t)NN * 4;
  float*    ald1 = ws + o; o += (size_t)NN * 4;
  unsigned* m1   = (unsigned*)(ws + o); o += (size_t)NN * 4;
  float*    den1 = ws + o; o += (size_t)NN * 4;
  float*    ew1  = ws + o; o += (size_t)EPLUS * 4;
  float*    als2 = ws + o; o += (size_t)NN;
  float*    ald2 = ws + o; o += (size_t)NN;
  unsigned* m2   = (unsigned*)(ws + o); o += (size_t)NN;
  float*    den2 = ws + o; o += (size_t)NN;
  float*    ew2  = ws + o; o += (size_t)EPLUS;
  float*    gsum = ws + o; o += (size_t)GRAPHS * 64;
  float*    gcnt = ws + o; o += (size_t)GRAPHS;

  // ---- init accumulators (ws/d_out are poisoned by harness) ----
  fill_u32_kernel<<<cdivll((long long)NN * 256, 256), 256, 0, stream>>>((unsigned*)out1, 0u, NN * 256);
  fill_u32_kernel<<<cdivll((long long)NN * 64, 256), 256, 0, stream>>>((unsigned*)out2, 0u, NN * 64);
  fill_u32_kernel<<<cdivll((long long)NN * 4, 256), 256, 0, stream>>>(m1, ENC_NEG_INF, NN * 4);
  fill_u32_kernel<<<cdivll((long long)NN * 4, 256), 256, 0, stream>>>((unsigned*)den1, 0u, NN * 4);
  fill_u32_kernel<<<cdivll((long long)NN, 256), 256, 0, stream>>>(m2, ENC_NEG_INF, NN);
  fill_u32_kernel<<<cdivll((long long)NN, 256), 256, 0, stream>>>((unsigned*)den2, 0u, NN);
  fill_u32_kernel<<<cdivll(GRAPHS * 64 + GRAPHS, 256), 256, 0, stream>>>((unsigned*)gsum, 0u,
                                                                        GRAPHS * 64 + GRAPHS);

  // ---- layer 1: h1 = x @ W1  (3125 x 16 tiles, 4 waves/block) ----
  wmma_gemm_f32<64, 256><<<(3125 * 16) / 4, 128, 0, stream>>>(x, W1, h1);
  attn_dots_kernel<<<cdivll((long long)NN * 4, 256), 256, 0, stream>>>(h1, a_src1, a_dst1, als1, ald1, 4);
  edge_max_kernel<4><<<cdivll((long long)EPLUS * 4, 256), 256, 0, stream>>>(als1, ald1, ei, m1);
  edge_expsum_kernel<4><<<cdivll((long long)EPLUS * 4, 256), 256, 0, stream>>>(als1, ald1, ei, m1, den1, ew1);
  edge_agg_kernel<4, 64><<<EPLUS, 256, 0, stream>>>(h1, ew1, den1, ei, out1);
  bias_elu_kernel<<<cdivll((long long)NN * 256, 256), 256, 0, stream>>>(out1, b1);

  // ---- layer 2: h2 = out1 @ W2  (3125 x 4 tiles) ----
  wmma_gemm_f32<256, 64><<<(3125 * 4) / 4, 128, 0, stream>>>(out1, W2, h2);
  attn_dots_kernel<<<cdivll((long long)NN, 256), 256, 0, stream>>>(h2, a_src2, a_dst2, als2, ald2, 1);
  edge_max_kernel<1><<<cdivll((long long)EPLUS, 256), 256, 0, stream>>>(als2, ald2, ei, m2);
  edge_expsum_kernel<1><<<cdivll((long long)EPLUS, 256), 256, 0, stream>>>(als2, ald2, ei, m2, den2, ew2);
  edge_agg_kernel<1, 64><<<EPLUS, 64, 0, stream>>>(h2, ew2, den2, ei, out2);

  // ---- pool + classifier ----
  pool_kernel<<<cdivll((long long)NN * 64, 256), 256, 0, stream>>>(out2, b2, batch, gsum, gcnt);
  classifier_kernel<<<cdivll(GRAPHS * NCLS, 256), 256, 0, stream>>>(gsum, gcnt, Wc, bc, (float*)d_out);
}